// EdgeConcatAdjNet_56805237457601
// MI455X (gfx1250) — compile-verified
//
#include <hip/hip_runtime.h>

// ---------------------------------------------------------------------------
// EdgeConcatAdjNet fused kernel for MI455X (gfx1250, wave32, WMMA).
//
// out[b,i,j] = relu(relu(pj[b,j]+pi[b,i]+b1) @ W2 + b2) @ W3 + b3
// Dominant GEMM: M = B*N*N = 2M "pairs", K = H1 = 64, N = H2 = 32 (fp32).
// Never materialize h (would be 512 MB) -> fuse with V_WMMA_F32_16X16X4_F32.
// adj and mask inputs are unused by the reference and are ignored.
// ---------------------------------------------------------------------------

typedef __attribute__((ext_vector_type(2))) float v2f;
typedef __attribute__((ext_vector_type(8))) float v8f;

#define B_    2
#define N_    1024
#define F_    128
#define H1_   64
#define H2_   32
#define WAVES 8     // waves per block
#define IPW   4     // i-rows per wave per block

// workspace layout (floats)
//   PJL : B * (N/16) * 1024   = 131072   (pj + b1, pre-shuffled WMMA-A layout)
//   PI  : B * N * 64          = 131072   (pi, plain row-major)
//   W2L : 2048                           (W2 pre-shuffled WMMA-B layout)
#define WS_PJL 0
#define WS_PI  131072
#define WS_W2L 262144

// --------------------------------------------------------------------------
// Kernel 1: projections. One block (64 threads) per (b,n) row.
// Thread t owns output channel k=t for both pj' (=pj+b1) and pi.
// pj' is scattered into the WMMA A-matrix lane layout per 16-row j-tile:
//   idx = kk*64 + r*32 + half*16 + m   with k = 4*kk + 2*half + r, m = n&15
// --------------------------------------------------------------------------
__global__ void __launch_bounds__(64) proj_kernel(
    const float* __restrict__ x, const float* __restrict__ W1,
    const float* __restrict__ b1, float* __restrict__ PJL,
    float* __restrict__ PI) {
  __shared__ float xs[F_];
  const int row = blockIdx.x;          // 0 .. B*N-1
  const int b   = row / N_;
  const int n   = row % N_;
  const int t   = threadIdx.x;         // k index, 0..63
  for (int f = t; f < F_; f += 64) xs[f] = x[row * F_ + f];
  __syncthreads();
  float accj = b1[t];
  float acci = 0.0f;
#pragma unroll 8
  for (int f = 0; f < F_; ++f) {
    const float xv = xs[f];
    accj = fmaf(xv, W1[f * H1_ + t], accj);          // Wj = W1[:F]
    acci = fmaf(xv, W1[(F_ + f) * H1_ + t], acci);   // Wi = W1[F:]
  }
  PI[row * H1_ + t] = acci;
  const int kk = t >> 2, half = (t >> 1) & 1, r = t & 1;
  const int jt = n >> 4, m = n & 15;
  PJL[(b * (N_ / 16) + jt) * 1024 + kk * 64 + r * 32 + half * 16 + m] = accj;
}

// --------------------------------------------------------------------------
// Kernel 2: shuffle W2 (64x32 row-major) into WMMA B-matrix lane layout.
//   W2L[nt][kk][r][half][n] = W2[4*kk + 2*half + r][nt*16 + n]
// --------------------------------------------------------------------------
__global__ void __launch_bounds__(256) w2shuf_kernel(
    const float* __restrict__ W2, float* __restrict__ W2L) {
  const int idx = blockIdx.x * 256 + threadIdx.x;
  if (idx < 2048) {
    const int nt = idx >> 10, kk = (idx >> 6) & 15;
    const int r = (idx >> 5) & 1, half = (idx >> 4) & 1, n = idx & 15;
    W2L[idx] = W2[(4 * kk + 2 * half + r) * H2_ + nt * 16 + n];
  }
}

// --------------------------------------------------------------------------
// Kernel 3: fused pairwise MLP.
// Block = 8 waves, grid = (j-tiles, i-groups of 32, B).
// Wave w handles i = ig*32 + w*IPW + ii for ii in [0,IPW); M-dim = 16 j's.
// Per 16 pairs: 16 K-steps x 2 N-tiles = 32 v_wmma_f32_16x16x4_f32.
// Epilogue fuses relu(.+b2) @ W3 + b3 with shfl_xor lane reductions.
// --------------------------------------------------------------------------
__global__ void __launch_bounds__(256) edge_kernel(
    const float* __restrict__ PJL, const float* __restrict__ PI,
    const float* __restrict__ W2L, const float* __restrict__ b2,
    const float* __restrict__ W3, const float* __restrict__ b3,
    float* __restrict__ out) {
  __shared__ float PJs[1024];             // A-layout pj' tile (16 j x 64 k)
  __shared__ float W2s[2048];             // B-layout W2 (2 n-tiles)
  __shared__ float PIs[WAVES * IPW * H1_];// 32 pi rows

  const int jt  = blockIdx.x;             // 0..63
  const int ig  = blockIdx.y;             // 0..31
  const int b   = blockIdx.z;
  const int tid = threadIdx.x;

  {
    const float* src = PJL + (b * (N_ / 16) + jt) * 1024;
    for (int idx = tid; idx < 1024; idx += 256) PJs[idx] = src[idx];
    for (int idx = tid; idx < 2048; idx += 256) W2s[idx] = W2L[idx];
    const float* psrc = PI + (b * N_ + ig * (WAVES * IPW)) * H1_;
    for (int idx = tid; idx < WAVES * IPW * H1_; idx += 256) PIs[idx] = psrc[idx];
  }
  __syncthreads();

  const int w    = tid >> 5;
  const int l    = tid & 31;
  const int lm   = l & 15;                // A: M-row / B: N-col for this lane
  const int half = l >> 4;
  const float b2v0 = b2[lm], b2v1 = b2[lm + 16];
  const float w3v0 = W3[lm], w3v1 = W3[lm + 16];
  const float b3v  = b3[0];
  const int j0 = jt * 16;

  for (int ii = 0; ii < IPW; ++ii) {
    const int il = w * IPW + ii;                  // local i (0..31)
    const int i  = ig * (WAVES * IPW) + il;

    // Per-lane pi registers: piL[kk*2+r] = pi[i][4*kk + 2*half + r]
    float piL[32];
#pragma unroll
    for (int t = 0; t < 32; ++t) {
      const int kk = t >> 1, r = t & 1;
      piL[t] = PIs[il * H1_ + 4 * kk + 2 * half + r];
    }

    v8f c0 = {}; v8f c1 = {};
#pragma unroll
    for (int kk = 0; kk < 16; ++kk) {
      // A-tile element (on the fly): h = relu(pj' + pi)
      const float a0 = fmaxf(PJs[kk * 64 +      half * 16 + lm] + piL[2 * kk    ], 0.0f);
      const float a1 = fmaxf(PJs[kk * 64 + 32 + half * 16 + lm] + piL[2 * kk + 1], 0.0f);
      v2f A;  A.x  = a0; A.y = a1;
      v2f B0; B0.x = W2s[kk * 64 +      half * 16 + lm];
              B0.y = W2s[kk * 64 + 32 + half * 16 + lm];
      v2f B1; B1.x = W2s[1024 + kk * 64 +      half * 16 + lm];
              B1.y = W2s[1024 + kk * 64 + 32 + half * 16 + lm];
      c0 = __builtin_amdgcn_wmma_f32_16x16x4_f32(false, A, false, B0,
                                                 (short)0, c0, false, false);
      c1 = __builtin_amdgcn_wmma_f32_16x16x4_f32(false, A, false, B1,
                                                 (short)0, c1, false, false);
    }

    // Epilogue: res[m] = sum_n relu(c[m][n]+b2[n]) * W3[n]  (n = 0..31)
    float res[8];
#pragma unroll
    for (int r = 0; r < 8; ++r) {
      float t0 = fmaxf(c0[r] + b2v0, 0.0f) * w3v0 +
                 fmaxf(c1[r] + b2v1, 0.0f) * w3v1;
      t0 += __shfl_xor(t0, 1, 32);
      t0 += __shfl_xor(t0, 2, 32);
      t0 += __shfl_xor(t0, 4, 32);
      t0 += __shfl_xor(t0, 8, 32);        // reduce over the 16 lanes of a half
      res[r] = t0 + b3v;
    }
    if (lm == 0) {                        // lanes 0 and 16 write 8 floats each
      float* dst = out + (size_t)(b * N_ + i) * N_ + j0 + 8 * half;
#pragma unroll
      for (int r = 0; r < 8; ++r) dst[r] = res[r];  // D: M = r + 8*half
    }
  }
}

extern "C" void kernel_launch(void* const* d_in, const int* in_sizes, int n_in,
                              void* d_out, int out_size, void* d_ws, size_t ws_size,
                              hipStream_t stream) {
  const float* x  = (const float*)d_in[0];
  // d_in[1] = adj (unused by reference), d_in[2] = mask (unused by reference)
  const float* W1 = (const float*)d_in[3];
  const float* b1 = (const float*)d_in[4];
  const float* W2 = (const float*)d_in[5];
  const float* b2 = (const float*)d_in[6];
  const float* W3 = (const float*)d_in[7];
  const float* b3 = (const float*)d_in[8];
  float* out = (float*)d_out;

  float* ws  = (float*)d_ws;
  float* PJL = ws + WS_PJL;
  float* PI  = ws + WS_PI;
  float* W2L = ws + WS_W2L;

  proj_kernel<<<B_ * N_, 64, 0, stream>>>(x, W1, b1, PJL, PI);
  w2shuf_kernel<<<8, 256, 0, stream>>>(W2, W2L);
  edge_kernel<<<dim3(N_ / 16, N_ / (WAVES * IPW), B_), 256, 0, stream>>>(
      PJL, PI, W2L, b2, W3, b3, out);
}